// Attention_73701638800162
// MI455X (gfx1250) — compile-verified
//
#include <hip/hip_runtime.h>
#include <hip/hip_bf16.h>
#include <math.h>

typedef __attribute__((ext_vector_type(2))) float v2f;
typedef __attribute__((ext_vector_type(8))) float v8f;

#define HH 1024
#define WW 1024
#define NPIX (HH * WW)
#define NBATCH 4

// ---------------------------------------------------------------------------
// ws layout (floats):
//   [0, 64)   : acc[b][16]  -> 0..2 Sqq[c], 3..5 Skk[d], 6..14 Sqk[c*3+d]
//   [64, 128) : M[b][16]    -> 0..8 fused (proj_w @ attn)[c*3+d]
// ---------------------------------------------------------------------------

__global__ void zero_acc_kernel(float* __restrict__ acc) {
  int i = threadIdx.x;
  if (i < NBATCH * 16) acc[i] = 0.f;
}

__device__ __forceinline__ float waveReduceSum(float v) {
#pragma unroll
  for (int off = 16; off > 0; off >>= 1) v += __shfl_xor(v, off, 32);
  return v;
}

// ---------------------------------------------------------------------------
// Pass 1: q = dw3x3(conv1x1(fhigh)), k = dw3x3(conv1x1(x)); accumulate the 15
// per-batch reduction scalars. 32x32 tile per 256-thread block (4 px/thread).
// ---------------------------------------------------------------------------
__global__ __launch_bounds__(256)
void conv_stats_kernel(const float* __restrict__ x, const float* __restrict__ fh,
                       const float* __restrict__ qCw, const float* __restrict__ qdw,
                       const float* __restrict__ kCw, const float* __restrict__ kdw,
                       float* __restrict__ acc) {
  __shared__ float sQ[3][34][34];  // conv1x1(fhigh) with 1-px halo
  __shared__ float sK[3][34][34];  // conv1x1(x)     with 1-px halo
  __shared__ float red[8][16];

  const int b = blockIdx.z;
  const int gx0 = blockIdx.x * 32 - 1;
  const int gy0 = blockIdx.y * 32 - 1;
  const int tid = threadIdx.y * 32 + threadIdx.x;

  float wq[9], wk[9], dq[27], dk[27];
#pragma unroll
  for (int i = 0; i < 9; ++i) { wq[i] = qCw[i]; wk[i] = kCw[i]; }
#pragma unroll
  for (int i = 0; i < 27; ++i) { dq[i] = qdw[i]; dk[i] = kdw[i]; }

  // Halo load + fused conv1x1 (channel mix) into LDS.
  for (int idx = tid; idx < 34 * 34; idx += 256) {
    const int ly = idx / 34, lx = idx - ly * 34;
    const int gy = gy0 + ly, gx = gx0 + lx;
    float f0 = 0.f, f1 = 0.f, f2 = 0.f, a0 = 0.f, a1 = 0.f, a2 = 0.f;
    if (gy >= 0 && gy < HH && gx >= 0 && gx < WW) {
      const size_t fo = ((size_t)b * 3 * HH + (size_t)gy) * WW + gx;
      f0 = fh[fo];
      f1 = fh[fo + (size_t)NPIX];
      f2 = fh[fo + 2 * (size_t)NPIX];
      const size_t xo = ((size_t)b * NPIX + (size_t)gy * WW + gx) * 3;
      a0 = x[xo]; a1 = x[xo + 1]; a2 = x[xo + 2];
    }
#pragma unroll
    for (int o = 0; o < 3; ++o) {
      sQ[o][ly][lx] = wq[o * 3 + 0] * f0 + wq[o * 3 + 1] * f1 + wq[o * 3 + 2] * f2;
      sK[o][ly][lx] = wk[o * 3 + 0] * a0 + wk[o * 3 + 1] * a1 + wk[o * 3 + 2] * a2;
    }
  }
  __syncthreads();

  float s[15];
#pragma unroll
  for (int i = 0; i < 15; ++i) s[i] = 0.f;

  const int tx = threadIdx.x;
#pragma unroll
  for (int py = 0; py < 4; ++py) {
    const int ty = threadIdx.y + py * 8;
    float q[3], k[3];
#pragma unroll
    for (int o = 0; o < 3; ++o) {
      float aq = 0.f, ak = 0.f;
#pragma unroll
      for (int ky = 0; ky < 3; ++ky)
#pragma unroll
        for (int kx = 0; kx < 3; ++kx) {
          aq = fmaf(dq[o * 9 + ky * 3 + kx], sQ[o][ty + ky][tx + kx], aq);
          ak = fmaf(dk[o * 9 + ky * 3 + kx], sK[o][ty + ky][tx + kx], ak);
        }
      q[o] = aq; k[o] = ak;
    }
#pragma unroll
    for (int c = 0; c < 3; ++c) { s[c] += q[c] * q[c]; s[3 + c] += k[c] * k[c]; }
#pragma unroll
    for (int c = 0; c < 3; ++c)
#pragma unroll
      for (int d = 0; d < 3; ++d) s[6 + c * 3 + d] += q[c] * k[d];
  }

  // wave32 tree reduce, then cross-wave reduce, then 15 global atomics per WG.
#pragma unroll
  for (int i = 0; i < 15; ++i) {
    const float v = waveReduceSum(s[i]);
    if (threadIdx.x == 0) red[threadIdx.y][i] = v;
  }
  __syncthreads();
  if (threadIdx.y == 0 && threadIdx.x < 15) {
    float v = 0.f;
#pragma unroll
    for (int w = 0; w < 8; ++w) v += red[w][threadIdx.x];
    atomicAdd(&acc[b * 16 + threadIdx.x], v);
  }
}

// ---------------------------------------------------------------------------
// Pass 2: per batch: norms -> 3x3 attn -> softmax -> M = proj_w @ attn via
// V_WMMA_F32_16X16X4_F32 (A = proj_w padded to 16x4, B = attn padded to 4x16).
// One wave per batch; operand build is branchless so EXEC is all 1s at WMMA.
// ---------------------------------------------------------------------------
__device__ __forceinline__ float mat3(const float m[9], int r, int c) {
  float v = 0.f;  // m[r][c] if r<3 && c<3, else 0 (select chain, no branches)
#pragma unroll
  for (int i = 0; i < 3; ++i)
#pragma unroll
    for (int j = 0; j < 3; ++j) v = (r == i && c == j) ? m[i * 3 + j] : v;
  return v;
}

__global__ __launch_bounds__(32)
void attn_mat_kernel(const float* __restrict__ acc, const float* __restrict__ pw,
                     const float* __restrict__ temp, float* __restrict__ Mout) {
  const int b = blockIdx.x;
  const int lane = threadIdx.x;
  const float* a = acc + b * 16;
  const float T = temp[0];

  float nq[3], nk[3];
#pragma unroll
  for (int c = 0; c < 3; ++c) {
    nq[c] = fmaxf(sqrtf(a[c]), 1e-12f);       // F.normalize eps semantics
    nk[c] = fmaxf(sqrtf(a[3 + c]), 1e-12f);
  }
  float AT[9];
#pragma unroll
  for (int c = 0; c < 3; ++c) {
    const float r0 = a[6 + c * 3 + 0] / (nq[c] * nk[0]) * T;
    const float r1 = a[6 + c * 3 + 1] / (nq[c] * nk[1]) * T;
    const float r2 = a[6 + c * 3 + 2] / (nq[c] * nk[2]) * T;
    const float m = fmaxf(r0, fmaxf(r1, r2));
    const float e0 = expf(r0 - m), e1 = expf(r1 - m), e2 = expf(r2 - m);
    const float inv = 1.f / (e0 + e1 + e2);
    AT[c * 3 + 0] = e0 * inv; AT[c * 3 + 1] = e1 * inv; AT[c * 3 + 2] = e2 * inv;
  }
  float P[9];
#pragma unroll
  for (int i = 0; i < 9; ++i) P[i] = pw[i];

  // f32 16x16x4 operand layout: A vgpr0 holds K={0,2}, vgpr1 K={1,3} split on
  // lane halves; B symmetric over its rows. D: vgpr r = row r (lanes 0-15).
  const int half = lane >> 4;
  const int r = lane & 15;
  v2f Av, Bv;
  Av.x = mat3(P, r, 2 * half);
  Av.y = mat3(P, r, 2 * half + 1);
  Bv.x = mat3(AT, 2 * half, r);
  Bv.y = mat3(AT, 2 * half + 1, r);
  v8f Cv = {0.f, 0.f, 0.f, 0.f, 0.f, 0.f, 0.f, 0.f};
  v8f D = __builtin_amdgcn_wmma_f32_16x16x4_f32(false, Av, false, Bv,
                                                (short)0, Cv, false, false);
  if (lane < 3) {
    Mout[b * 16 + 0 * 3 + lane] = D[0];
    Mout[b * 16 + 1 * 3 + lane] = D[1];
    Mout[b * 16 + 2 * 3 + lane] = D[2];
  }
}

// ---------------------------------------------------------------------------
// Pass 3: recompute v = dw3x3(conv1x1(x)) (cheaper than storing 50MB), apply
// out_c = sum_d M[c,d] * v_d + bias_c, write channel-interleaved [B,N,3].
// ---------------------------------------------------------------------------
__global__ __launch_bounds__(256)
void apply_kernel(const float* __restrict__ x, const float* __restrict__ kCw,
                  const float* __restrict__ kdw, const float* __restrict__ Mws,
                  const float* __restrict__ pb, float* __restrict__ out) {
  __shared__ float sK[3][34][34];
  const int b = blockIdx.z;
  const int gx0 = blockIdx.x * 32 - 1;
  const int gy0 = blockIdx.y * 32 - 1;
  const int tid = threadIdx.y * 32 + threadIdx.x;

  float wk[9], dk[27], M[9];
#pragma unroll
  for (int i = 0; i < 9; ++i) { wk[i] = kCw[i]; M[i] = Mws[b * 16 + i]; }
#pragma unroll
  for (int i = 0; i < 27; ++i) dk[i] = kdw[i];
  const float pb0 = pb[0], pb1 = pb[1], pb2 = pb[2];

  for (int idx = tid; idx < 34 * 34; idx += 256) {
    const int ly = idx / 34, lx = idx - ly * 34;
    const int gy = gy0 + ly, gx = gx0 + lx;
    float a0 = 0.f, a1 = 0.f, a2 = 0.f;
    if (gy >= 0 && gy < HH && gx >= 0 && gx < WW) {
      const size_t xo = ((size_t)b * NPIX + (size_t)gy * WW + gx) * 3;
      a0 = x[xo]; a1 = x[xo + 1]; a2 = x[xo + 2];
    }
#pragma unroll
    for (int o = 0; o < 3; ++o)
      sK[o][ly][lx] = wk[o * 3 + 0] * a0 + wk[o * 3 + 1] * a1 + wk[o * 3 + 2] * a2;
  }
  __syncthreads();

  const int tx = threadIdx.x;
#pragma unroll
  for (int py = 0; py < 4; ++py) {
    const int ty = threadIdx.y + py * 8;
    float k[3];
#pragma unroll
    for (int o = 0; o < 3; ++o) {
      float ak = 0.f;
#pragma unroll
      for (int ky = 0; ky < 3; ++ky)
#pragma unroll
        for (int kx = 0; kx < 3; ++kx)
          ak = fmaf(dk[o * 9 + ky * 3 + kx], sK[o][ty + ky][tx + kx], ak);
      k[o] = ak;
    }
    const int gy = blockIdx.y * 32 + ty;
    const int gx = blockIdx.x * 32 + tx;
    const size_t oo = ((size_t)b * NPIX + (size_t)gy * WW + gx) * 3;
    out[oo + 0] = fmaf(M[0], k[0], fmaf(M[1], k[1], fmaf(M[2], k[2], pb0)));
    out[oo + 1] = fmaf(M[3], k[0], fmaf(M[4], k[1], fmaf(M[5], k[2], pb1)));
    out[oo + 2] = fmaf(M[6], k[0], fmaf(M[7], k[1], fmaf(M[8], k[2], pb2)));
  }
}

extern "C" void kernel_launch(void* const* d_in, const int* in_sizes, int n_in,
                              void* d_out, int out_size, void* d_ws, size_t ws_size,
                              hipStream_t stream) {
  (void)in_sizes; (void)n_in; (void)out_size; (void)ws_size;
  const float* x    = (const float*)d_in[0];  // [B, N, 3]
  const float* fh   = (const float*)d_in[1];  // [B, 3, H, W]
  const float* qCw  = (const float*)d_in[2];  // [3,3,1,1]
  const float* qdw  = (const float*)d_in[3];  // [3,1,3,3]
  const float* kCw  = (const float*)d_in[4];  // [3,3,1,1]
  const float* kdw  = (const float*)d_in[5];  // [3,1,3,3]
  const float* pw   = (const float*)d_in[6];  // [3,3,1,1]
  const float* pb   = (const float*)d_in[7];  // [3]
  const float* temp = (const float*)d_in[8];  // [1,1,1]
  float* out = (float*)d_out;

  float* acc = (float*)d_ws;      // NBATCH*16 floats
  float* M   = acc + 64;          // NBATCH*16 floats

  zero_acc_kernel<<<1, 64, 0, stream>>>(acc);

  dim3 blk(32, 8);
  dim3 grd(WW / 32, HH / 32, NBATCH);
  conv_stats_kernel<<<grd, blk, 0, stream>>>(x, fh, qCw, qdw, kCw, kdw, acc);
  attn_mat_kernel<<<NBATCH, 32, 0, stream>>>(acc, pw, temp, M);
  apply_kernel<<<grd, blk, 0, stream>>>(x, kCw, kdw, M, pb, out);
}